// GCNModel_31610959298975
// MI455X (gfx1250) — compile-verified
//
#include <hip/hip_runtime.h>

typedef __attribute__((ext_vector_type(2))) float v2f;
typedef __attribute__((ext_vector_type(8))) float v8f;

#define FDIM 128   // F_IN == H == 128

// ---------------- utility kernels ----------------
__global__ __launch_bounds__(256) void k_init(int* __restrict__ deg,
                                              float* __restrict__ stats, int N)
{
    int i = blockIdx.x * 256 + threadIdx.x;
    if (i < N) deg[i] = 1;           // self-loop contribution to degree
    if (i < 256) stats[i] = 0.f;     // gsum[128] + gsq[128]
}

__global__ __launch_bounds__(256) void k_deg(const int* __restrict__ dst,
                                             int* __restrict__ deg, int E)
{
    int i = blockIdx.x * 256 + threadIdx.x;
    if (i < E) atomicAdd(&deg[dst[i]], 1);
}

__global__ __launch_bounds__(256) void k_dis(const int* __restrict__ deg,
                                             float* __restrict__ dis, int N)
{
    int i = blockIdx.x * 256 + threadIdx.x;
    if (i < N) dis[i] = rsqrtf((float)deg[i]);
}

// ---------------- GEMM1: h1 = x @ W1  (N x 128) @ (128 x 128), f32 WMMA ----
// Block: 256 thr = 8 waves = 2 row-tiles x 4 col-tiles (32 rows x 64 cols).
__global__ __launch_bounds__(256) void k_gemm1(const float* __restrict__ X,
                                               const float* __restrict__ W,
                                               float* __restrict__ O, int N)
{
    __shared__ float lA[32 * FDIM];   // 16 KB, row-major 32 x 128
    __shared__ v2f   lB[64 * 64];     // 32 KB, K-pair packed: [k/2][col]

    const int t    = threadIdx.x;
    const int rowb = blockIdx.x * 32;
    const int colb = blockIdx.y * 64;

    for (int idx = t; idx < 32 * FDIM; idx += 256) {
        int r = idx >> 7, k = idx & 127;
        int gr = rowb + r; if (gr >= N) gr = N - 1;
        lA[idx] = X[(size_t)gr * FDIM + k];
    }
    for (int idx = t; idx < 64 * 64; idx += 256) {
        int k2 = idx >> 6, c = idx & 63;
        const float* wp = W + (size_t)(2 * k2) * FDIM + (colb + c);
        v2f b; b.x = wp[0]; b.y = wp[FDIM];
        lB[idx] = b;
    }
    __syncthreads();

    const int wave = t >> 5, lane = t & 31;
    const int m = lane & 15, hi = lane >> 4;
    const int rt = wave >> 2, ct = wave & 3;
    const float* Arow = &lA[(rt * 16 + m) * FDIM];
    const int nloc = ct * 16 + m;

    v8f acc = {0.f, 0.f, 0.f, 0.f, 0.f, 0.f, 0.f, 0.f};
#pragma unroll
    for (int k = 0; k < FDIM; k += 4) {
        v2f a = *(const v2f*)&Arow[k + 2 * hi];          // K = k+2hi, k+2hi+1
        v2f b = lB[(k / 2 + hi) * 64 + nloc];            // mirrored layout
        acc = __builtin_amdgcn_wmma_f32_16x16x4_f32(
            false, a, false, b, (short)0, acc, false, false);
    }

    const int col = colb + nloc;
#pragma unroll
    for (int r = 0; r < 8; ++r) {
        int row = rowb + rt * 16 + r + 8 * hi;
        if (row < N) O[(size_t)row * FDIM + col] = acc[r];
    }
}

// ---------------- self-loop init (also initializes accumulator with self term + bias) ------
__global__ __launch_bounds__(256) void k_self128(const float* __restrict__ h,
                                                 const float* __restrict__ dis,
                                                 const float* __restrict__ bias,
                                                 float* __restrict__ out, int N)
{
    long long idx = (long long)blockIdx.x * 256 + threadIdx.x;
    if (idx >= (long long)N * FDIM) return;
    int i = (int)(idx >> 7), f = (int)(idx & 127);
    float d = dis[i];
    out[idx] = h[idx] * d * d + bias[f];
}

__global__ __launch_bounds__(256) void k_selfC(const float* __restrict__ h,
                                               const float* __restrict__ dis,
                                               const float* __restrict__ bias,
                                               float* __restrict__ out, int N, int C)
{
    long long idx = (long long)blockIdx.x * 256 + threadIdx.x;
    if (idx >= (long long)N * C) return;
    int i = (int)(idx / C), f = (int)(idx - (long long)i * C);
    float d = dis[i];
    out[idx] = h[idx] * d * d + bias[f];
}

// ---------------- edge scatter-add, one wave per edge ---------------------
__global__ __launch_bounds__(256) void k_edge128(const float* __restrict__ h,
                                                 const int* __restrict__ src,
                                                 const int* __restrict__ dst,
                                                 const float* __restrict__ dis,
                                                 float* __restrict__ out, int E)
{
    long long gid = (long long)blockIdx.x * 256 + threadIdx.x;
    int e = (int)(gid >> 5);
    int lane = threadIdx.x & 31;
    if (e >= E) return;
    int s = src[e], d = dst[e];
    float w = dis[s] * dis[d];
    const float4* hp = (const float4*)(h + (size_t)s * FDIM);
    float4 v = hp[lane];
    float* op = out + (size_t)d * FDIM + lane * 4;
    atomicAdd(op + 0, v.x * w);
    atomicAdd(op + 1, v.y * w);
    atomicAdd(op + 2, v.z * w);
    atomicAdd(op + 3, v.w * w);
}

__global__ __launch_bounds__(256) void k_edgeC(const float* __restrict__ h,
                                               const int* __restrict__ src,
                                               const int* __restrict__ dst,
                                               const float* __restrict__ dis,
                                               float* __restrict__ out, int E, int C)
{
    long long gid = (long long)blockIdx.x * 256 + threadIdx.x;
    int e = (int)(gid >> 5);
    int lane = threadIdx.x & 31;
    if (e >= E) return;
    int s = src[e], d = dst[e];
    float w = dis[s] * dis[d];
    const float* hp = h + (size_t)s * C;
    float* op = out + (size_t)d * C;
    if (lane < C)      atomicAdd(op + lane,      hp[lane]      * w);
    if (32 + lane < C) atomicAdd(op + 32 + lane, hp[32 + lane] * w);
}

// ---------------- ReLU in place + per-feature sum/sumsq -------------------
__global__ __launch_bounds__(256) void k_relu_stats(float* __restrict__ a,
                                                    float* __restrict__ gsum,
                                                    float* __restrict__ gsq, int N)
{
    __shared__ float ssum[256];
    __shared__ float ssq[256];
    int t = threadIdx.x, f = t & 127, sub = t >> 7;
    float sum = 0.f, sq = 0.f;
    int base = blockIdx.x * 64;
#pragma unroll 4
    for (int i = 0; i < 32; ++i) {
        int row = base + sub + 2 * i;
        if (row < N) {
            size_t idx = (size_t)row * FDIM + f;
            float v = a[idx];
            v = fmaxf(v, 0.f);
            a[idx] = v;
            sum += v;
            sq  += v * v;
        }
    }
    ssum[t] = sum; ssq[t] = sq;
    __syncthreads();
    if (t < 128) {
        atomicAdd(&gsum[t], ssum[t] + ssum[t + 128]);
        atomicAdd(&gsq[t],  ssq[t]  + ssq[t + 128]);
    }
}

__global__ void k_bn_finalize(const float* __restrict__ gsum,
                              const float* __restrict__ gsq,
                              const float* __restrict__ gamma,
                              const float* __restrict__ beta,
                              float* __restrict__ scl,
                              float* __restrict__ sft, int N)
{
    int f = threadIdx.x;
    float inv_n = 1.f / (float)N;
    float mean = gsum[f] * inv_n;
    float var  = gsq[f] * inv_n - mean * mean;
    float inv  = rsqrtf(var + 1e-5f);
    float g = gamma[f] * inv;
    scl[f] = g;
    sft[f] = beta[f] - mean * g;
}

// ---------------- GEMM2: h2 = BN(agg1) @ W2  (128 -> C<=48), f32 WMMA -----
// Block: 96 thr = 3 waves = 3 col-tiles (48 cols, guarded), one 16-row tile.
__global__ __launch_bounds__(96) void k_gemm2(const float* __restrict__ A,
                                              const float* __restrict__ W,
                                              const float* __restrict__ scl,
                                              const float* __restrict__ sft,
                                              float* __restrict__ O, int N, int C)
{
    __shared__ float lA[16 * FDIM];  // 8 KB (BN applied during staging)
    __shared__ v2f   lB[64 * 48];    // 24 KB, K-pair packed, cols padded to 48

    const int t = threadIdx.x;
    const int rowb = blockIdx.x * 16;

    for (int idx = t; idx < 16 * FDIM; idx += 96) {
        int r = idx >> 7, k = idx & 127;
        int gr = rowb + r; if (gr >= N) gr = N - 1;
        lA[idx] = A[(size_t)gr * FDIM + k] * scl[k] + sft[k];
    }
    for (int idx = t; idx < 64 * 48; idx += 96) {
        int k2 = idx / 48, c = idx - k2 * 48;
        v2f b;
        b.x = (c < C) ? W[(size_t)(2 * k2) * C + c]     : 0.f;
        b.y = (c < C) ? W[(size_t)(2 * k2 + 1) * C + c] : 0.f;
        lB[idx] = b;
    }
    __syncthreads();

    const int wave = t >> 5, lane = t & 31;
    const int m = lane & 15, hi = lane >> 4;
    const float* Arow = &lA[m * FDIM];
    const int nloc = wave * 16 + m;

    v8f acc = {0.f, 0.f, 0.f, 0.f, 0.f, 0.f, 0.f, 0.f};
#pragma unroll
    for (int k = 0; k < FDIM; k += 4) {
        v2f a = *(const v2f*)&Arow[k + 2 * hi];
        v2f b = lB[(k / 2 + hi) * 48 + nloc];
        acc = __builtin_amdgcn_wmma_f32_16x16x4_f32(
            false, a, false, b, (short)0, acc, false, false);
    }

    if (nloc < C) {
#pragma unroll
        for (int r = 0; r < 8; ++r) {
            int row = rowb + r + 8 * hi;
            if (row < N) O[(size_t)row * C + nloc] = acc[r];
        }
    }
}

// ---------------- log-softmax over C<=64, one wave per row ----------------
__global__ __launch_bounds__(256) void k_logsoftmax(float* __restrict__ out, int N, int C)
{
    long long gid = (long long)blockIdx.x * 256 + threadIdx.x;
    int row = (int)(gid >> 5);
    int lane = threadIdx.x & 31;
    if (row >= N) return;
    float* p = out + (size_t)row * C;
    float v0 = (lane < C) ? p[lane] : -3.4e38f;
    float v1 = (32 + lane < C) ? p[32 + lane] : -3.4e38f;
    float mx = fmaxf(v0, v1);
#pragma unroll
    for (int o = 16; o > 0; o >>= 1) mx = fmaxf(mx, __shfl_xor(mx, o, 32));
    float s = ((lane < C) ? expf(v0 - mx) : 0.f) +
              ((32 + lane < C) ? expf(v1 - mx) : 0.f);
#pragma unroll
    for (int o = 16; o > 0; o >>= 1) s += __shfl_xor(s, o, 32);
    float l = logf(s);
    if (lane < C)      p[lane]      = v0 - mx - l;
    if (32 + lane < C) p[32 + lane] = v1 - mx - l;
}

// ---------------- host-side orchestration --------------------------------
extern "C" void kernel_launch(void* const* d_in, const int* in_sizes, int n_in,
                              void* d_out, int out_size, void* d_ws, size_t ws_size,
                              hipStream_t stream)
{
    const float* x     = (const float*)d_in[0];
    const int*   ei    = (const int*)  d_in[1];
    const float* W1    = (const float*)d_in[2];
    const float* b1    = (const float*)d_in[3];
    const float* gamma = (const float*)d_in[4];
    const float* beta  = (const float*)d_in[5];
    const float* W2    = (const float*)d_in[6];
    const float* b2    = (const float*)d_in[7];
    float* out = (float*)d_out;

    const int N = in_sizes[0] / FDIM;
    const int E = in_sizes[1] / 2;
    const int C = in_sizes[6] / FDIM;   // 40
    const int* src = ei;
    const int* dst = ei + E;

    // workspace layout
    char* ws = (char*)d_ws;
    size_t off = 0;
    auto up = [](size_t v) { return (v + 255) & ~(size_t)255; };
    int*   deg   = (int*)(ws + off);   off = up(off + (size_t)N * 4);
    float* dis   = (float*)(ws + off); off = up(off + (size_t)N * 4);
    float* stats = (float*)(ws + off); off = up(off + 512 * 4);
    float* gsum = stats, *gsq = stats + 128, *scl = stats + 256, *sft = stats + 384;
    float* h1   = (float*)(ws + off);  off = up(off + (size_t)N * FDIM * 4);
    float* agg1 = (float*)(ws + off);  off = up(off + (size_t)N * FDIM * 4);
    float* h2   = h1;  // h1 is dead after edge scatter #1; reuse for N x C

    const int nb_N   = (N + 255) / 256;
    const int nb_E   = (E + 255) / 256;
    const int nb_Ew  = (E + 7) / 8;               // one wave per edge
    const int nb_Nw  = (N + 7) / 8;               // one wave per row
    const long long nfe1 = (long long)N * FDIM;
    const long long nfe2 = (long long)N * C;

    // 1-3: degree, dis = rsqrt(deg)
    k_init<<<nb_N, 256, 0, stream>>>(deg, stats, N);
    k_deg<<<nb_E, 256, 0, stream>>>(dst, deg, E);
    k_dis<<<nb_N, 256, 0, stream>>>(deg, dis, N);

    // 4: h1 = x @ W1 (WMMA f32)
    k_gemm1<<<dim3((N + 31) / 32, FDIM / 64), 256, 0, stream>>>(x, W1, h1, N);

    // 5-6: conv1 aggregation (self-loop init + edge atomics)
    k_self128<<<(int)((nfe1 + 255) / 256), 256, 0, stream>>>(h1, dis, b1, agg1, N);
    k_edge128<<<nb_Ew, 256, 0, stream>>>(h1, src, dst, dis, agg1, E);

    // 7-8: ReLU + batchnorm statistics -> scale/shift
    k_relu_stats<<<(N + 63) / 64, 256, 0, stream>>>(agg1, gsum, gsq, N);
    k_bn_finalize<<<1, 128, 0, stream>>>(gsum, gsq, gamma, beta, scl, sft, N);

    // 9: h2 = BN(agg1) @ W2 (WMMA f32, BN fused into A staging)
    k_gemm2<<<(N + 15) / 16, 96, 0, stream>>>(agg1, W2, scl, sft, h2, N, C);

    // 10-11: conv2 aggregation into d_out
    k_selfC<<<(int)((nfe2 + 255) / 256), 256, 0, stream>>>(h2, dis, b2, out, N, C);
    k_edgeC<<<nb_Ew, 256, 0, stream>>>(h2, src, dst, dis, out, E, C);

    // 12: log-softmax in place
    k_logsoftmax<<<nb_Nw, 256, 0, stream>>>(out, N, C);
}